// ROIHead_57071525429516
// MI455X (gfx1250) — compile-verified
//
#include <hip/hip_runtime.h>
#include <stdint.h>

// ROI head proposal<->GT matching for MI455X (gfx1250, wave32).
// N = 200000 proposals, G = 128 GT boxes.
// Memory floor ~8 MB / 23.3 TB/s ~ 0.35 us; VALU-bound inner loop
// (min/max/clamp — not WMMA-expressible), so we minimize VALU+DS ops.
// GT table is staged into LDS via the CDNA5 async-to-LDS DMA path (ASYNCcnt).

#define GT_CAP 128

// gfx1250 async-to-LDS builtin takes int4 pointers: AS1 (global) src, AS3 (LDS) dst.
typedef __attribute__((ext_vector_type(4))) int v4i;
typedef __attribute__((address_space(1))) v4i* as1_v4ip;
typedef __attribute__((address_space(3))) v4i* as3_v4ip;

// Compare candidate IoU = inter/uni against best bi/bu without dividing:
// both unions are > 0, so inter/uni > bi/bu  <=>  inter*bu > bi*uni.
#define IOU_STEP(GX1, GY1, GX2, GY2, GA, GIDX)                          \
    {                                                                   \
        float ix1 = fmaxf(px1, (GX1));                                  \
        float iy1 = fmaxf(py1, (GY1));                                  \
        float ix2 = fminf(px2, (GX2));                                  \
        float iy2 = fminf(py2, (GY2));                                  \
        float iw  = fmaxf(ix2 - ix1, 0.0f);                             \
        float ih  = fmaxf(iy2 - iy1, 0.0f);                             \
        float inter = iw * ih;                                          \
        float uni   = parea + (GA) - inter;                             \
        if (inter * bu > bi * uni) { bi = inter; bu = uni; bidx = (GIDX); } \
    }

__global__ __launch_bounds__(256) void roi_assign_kernel(
    const float4* __restrict__ proposals,   // [N] as float4 (x1,y1,x2,y2)
    const float4* __restrict__ gt_boxes,    // [G] as float4
    const int*    __restrict__ gt_labels,   // [G]
    long long*    __restrict__ out_labels,  // [N] int64
    float4*       __restrict__ out_boxes,   // [N] float4
    int N, int G)
{
    __shared__ __align__(16) float4 sbox[GT_CAP];   // AoS copy (for final gather)
    __shared__ __align__(16) float  sx1[GT_CAP];
    __shared__ __align__(16) float  sy1[GT_CAP];
    __shared__ __align__(16) float  sx2[GT_CAP];
    __shared__ __align__(16) float  sy2[GT_CAP];
    __shared__ __align__(16) float  sar[GT_CAP];

    const int t = threadIdx.x;
    const int i = blockIdx.x * blockDim.x + t;

    // Kick off the per-lane proposal load early (LOADcnt, independent of the
    // LDS staging below) so it overlaps the barrier.
    float4 p = make_float4(0.f, 0.f, 0.f, 0.f);
    if (i < N) p = proposals[i];

    // ---- Stage GT boxes into LDS via the CDNA5 async-to-LDS DMA path ----
    if (t < G) {
#if __has_builtin(__builtin_amdgcn_global_load_async_to_lds_b128)
        __builtin_amdgcn_global_load_async_to_lds_b128(
            (as1_v4ip)(gt_boxes + t), (as3_v4ip)(sbox + t),
            /*offset=*/0, /*cpol=*/0);
#else
        sbox[t] = gt_boxes[t];
#endif
    }
#if __has_builtin(__builtin_amdgcn_s_wait_asynccnt)
    __builtin_amdgcn_s_wait_asynccnt(0);
#else
    asm volatile("s_wait_asynccnt 0" ::: "memory");
#endif
    __syncthreads();

    // ---- Build SoA (+ precomputed areas) so the hot loop reads float4 runs ----
    if (t < GT_CAP) {
        if (t < G) {
            float4 b = sbox[t];
            sx1[t] = b.x; sy1[t] = b.y; sx2[t] = b.z; sy2[t] = b.w;
            sar[t] = (b.z - b.x) * (b.w - b.y);
        } else {
            // Padding that can never win: zero-size box far away -> inter = 0.
            sx1[t] = 3.0e8f; sy1[t] = 3.0e8f; sx2[t] = 3.0e8f; sy2[t] = 3.0e8f;
            sar[t] = 0.0f;
        }
    }
    __syncthreads();

    if (i >= N) return;

    const float px1 = p.x, py1 = p.y, px2 = p.z, py2 = p.w;
    const float parea = (px2 - px1) * (py2 - py1);

    // Best IoU tracked as a fraction (bi/bu); init so the first candidate
    // always wins (matches argmax semantics when every IoU is 0).
    float bi = -1.0f;
    float bu =  1.0f;
    int   bidx = 0;

    const float4* X1 = (const float4*)sx1;
    const float4* Y1 = (const float4*)sy1;
    const float4* X2 = (const float4*)sx2;
    const float4* Y2 = (const float4*)sy2;
    const float4* AR = (const float4*)sar;

    const int G4 = (G + 3) >> 2;
    for (int g4 = 0; g4 < G4; ++g4) {
        const float4 x1 = X1[g4];
        const float4 y1 = Y1[g4];
        const float4 x2 = X2[g4];
        const float4 y2 = Y2[g4];
        const float4 ar = AR[g4];
        const int gb = g4 << 2;
        IOU_STEP(x1.x, y1.x, x2.x, y2.x, ar.x, gb + 0);
        IOU_STEP(x1.y, y1.y, x2.y, y2.y, ar.y, gb + 1);
        IOU_STEP(x1.z, y1.z, x2.z, y2.z, ar.z, gb + 2);
        IOU_STEP(x1.w, y1.w, x2.w, y2.w, ar.w, gb + 3);
    }

    const float best = bi / bu;   // single divide per proposal (bu > 0)

    long long lab;
    if (best < 0.1f)       lab = -1;                               // ignored
    else if (best < 0.5f)  lab = 0;                                // background
    else                   lab = (long long)gt_labels[bidx];       // foreground

    out_labels[i] = lab;
    out_boxes[i]  = sbox[bidx];   // raw-argmax gather, matches torch clamp trick
}

extern "C" void kernel_launch(void* const* d_in, const int* in_sizes, int n_in,
                              void* d_out, int out_size, void* d_ws, size_t ws_size,
                              hipStream_t stream) {
    const float4* proposals = (const float4*)d_in[0];
    const float4* gt_boxes  = (const float4*)d_in[1];
    const int*    gt_labels = (const int*)d_in[2];

    const int N = in_sizes[0] / 4;
    int G = in_sizes[1] / 4;
    if (G > GT_CAP) G = GT_CAP;

    // Outputs concatenated in return order: int64 labels [N], then f32 boxes [N,4].
    long long* out_labels = (long long*)d_out;
    float4*    out_boxes  = (float4*)((char*)d_out + (size_t)N * sizeof(long long));

    const int block = 256;                 // 8 wave32s per workgroup
    const int grid  = (N + block - 1) / block;
    roi_assign_kernel<<<grid, block, 0, stream>>>(
        proposals, gt_boxes, gt_labels, out_labels, out_boxes, N, G);
}